// LIFNodeTriton_15633680958004
// MI455X (gfx1250) — compile-verified
//
#include <hip/hip_runtime.h>

// ---------------------------------------------------------------------------
// LIF neuron scan, (N=64, T=4, F=65536) f32. Pure streaming kernel:
// 128 MB of traffic, ~5.5 us floor at 23.3 TB/s. Data path uses the CDNA5
// async global->LDS engine (ASYNCcnt) with double buffering to keep deep
// memory-level parallelism without VGPR pressure. WMMA is intentionally not
// used: there is no contraction dimension in this op.
// ---------------------------------------------------------------------------

#define T_STEPS 4
#define DECAY   0.5f
#define VTH     1.0f
#define BLOCK   256
#define VEC     4
#define TILE_F  (BLOCK * VEC)   // 1024 floats per tile row (4 KB)

typedef __attribute__((ext_vector_type(4))) float f32x4;
typedef int v4i __attribute__((ext_vector_type(4)));

#if defined(__AMDGCN__) && __has_builtin(__builtin_amdgcn_global_load_async_to_lds_b128) && __has_builtin(__builtin_amdgcn_s_wait_asynccnt)
#define LIF_ASYNC 1
// Builtin signature (from compiler diagnostic): arg0 = v4i32 in global AS
// (printed as "__device__"), arg1 = v4i32 in LDS AS, then imm offset, imm cpol.
typedef __attribute__((address_space(1))) v4i* gv4i_ptr;
typedef __attribute__((address_space(3))) v4i* lv4i_ptr;
#else
#define LIF_ASYNC 0
#endif

__global__ __launch_bounds__(BLOCK) void lif_scan_kernel(
    const float* __restrict__ x, float* __restrict__ out, int F, int tiles) {
#if LIF_ASYNC
  // double-buffered staging: 2 buffers x 4 time rows x 4KB = 32 KB LDS
  __shared__ f32x4 buf[2][T_STEPS][BLOCK];
#endif
  const int    n       = blockIdx.y;
  const int    tid     = threadIdx.x;
  const size_t rowbase = (size_t)n * T_STEPS * (size_t)F;

#if LIF_ASYNC
  auto issue = [&](int tile, int pb) {
    const size_t g0 = rowbase + (size_t)tile * TILE_F + (size_t)tid * VEC;
#pragma unroll
    for (int t = 0; t < T_STEPS; ++t) {
      __builtin_amdgcn_global_load_async_to_lds_b128(
          (gv4i_ptr)(x + g0 + (size_t)t * F),
          (lv4i_ptr)&buf[pb][t][tid],
          /*offset=*/0, /*cpol=*/0);
    }
  };
  issue((int)blockIdx.x, 0);
#endif

  int pb = 0;
  for (int tile = (int)blockIdx.x; tile < tiles; tile += (int)gridDim.x, pb ^= 1) {
    const size_t g0 = rowbase + (size_t)tile * TILE_F + (size_t)tid * VEC;
    f32x4 r[T_STEPS];

#if LIF_ASYNC
    const int next = tile + (int)gridDim.x;
    if (next < tiles) {
      issue(next, pb ^ 1);
      // 4 loads of the *next* tile may stay in flight; ours (issued earlier,
      // async loads retire in order) are then guaranteed complete.
      __builtin_amdgcn_s_wait_asynccnt(T_STEPS);
    } else {
      __builtin_amdgcn_s_wait_asynccnt(0);
    }
    __asm__ __volatile__("" ::: "memory");  // keep LDS reads after the wait
#pragma unroll
    for (int t = 0; t < T_STEPS; ++t) r[t] = buf[pb][t][tid];
#else
#pragma unroll
    for (int t = 0; t < T_STEPS; ++t)
      r[t] = *(const f32x4*)(x + g0 + (size_t)t * F);
#endif

    // LIF temporal scan. DECAY = 0.5 (power of two) => mul+add is exact,
    // matching the reference bit-for-bit regardless of fma contraction.
    f32x4 v = {0.f, 0.f, 0.f, 0.f};
#pragma unroll
    for (int t = 0; t < T_STEPS; ++t) {
      v = v * DECAY + r[t];
      f32x4 sp;
#pragma unroll
      for (int c = 0; c < 4; ++c) sp[c] = (v[c] >= VTH) ? 1.0f : 0.0f;
      v = v - sp * VTH;  // soft reset
      *(f32x4*)(out + g0 + (size_t)t * F) = sp;
    }
  }
}

extern "C" void kernel_launch(void* const* d_in, const int* in_sizes, int n_in,
                              void* d_out, int out_size, void* d_ws, size_t ws_size,
                              hipStream_t stream) {
  (void)n_in; (void)out_size; (void)d_ws; (void)ws_size;
  const float* x   = (const float*)d_in[0];
  float*       out = (float*)d_out;

  const int F     = 65536;                 // feature dim (reference shape)
  const int B     = in_sizes[0] / F;       // 256 rows = N*T
  const int N     = B / T_STEPS;           // 64 neuron row-groups
  const int tiles = F / TILE_F;            // 64 tiles of 1024 floats

  int strip = tiles < 32 ? tiles : 32;     // 2048 blocks -> ~32 MB in flight
  dim3 grid(strip, N);
  lif_scan_kernel<<<grid, BLOCK, 0, stream>>>(x, out, F, tiles);
}